// MonoDepthNet_scnn_33311766347998
// MI455X (gfx1250) — compile-verified
//
#include <hip/hip_runtime.h>

// SCNN-style bidirectional message passing:
//   pass1: out1[w] = x[w]    + relu(conv9_H(out1[w-1], w_lr)),  out1[0]   = x[0]
//   pass2: out2[w] = out1[w] + relu(conv9_H(out2[w+1], w_rl)),  out2[W-1] = out1[W-1]
// One workgroup (2 waves) per batch; conv via chained V_WMMA_F32_16X16X4_F32.

#define KS    9
#define BATCH 32
#define HH    384
#define WW    1280

typedef __attribute__((ext_vector_type(2))) float v2f;
typedef __attribute__((ext_vector_type(8))) float v8f;

#define NBLK      20    // padded LDS stride (dwords) per 16-h block -> conflict-free N-stride reads
#define COLBUF_DW 688   // covers ext h in [-16, 528)

// slot for extended h index e in [-16, 528): blocks of 20 dwords, block (e>>4)+1, offset e&15.
// Slots 16..19 of each block are padding; halos land in real, pre-zeroed slots.
__device__ __forceinline__ int eslot(int e) { return ((e >> 4) + 1) * NBLK + (e & 15); }

// One scan pass. src/dst are per-batch [H, W] fp32 planes. dir = +1 (L->R) or -1 (R->L).
__device__ __forceinline__ void run_pass(const float* __restrict__ src,
                                         float* __restrict__ dst,
                                         const float* __restrict__ w9g,
                                         float* __restrict__ cb0,
                                         float* __restrict__ cb1,
                                         int wave_base, int n, int hi,
                                         int h0, bool valid, int dir)
{
    // ---- banded A matrices (16x4 chunks of the 16x24 conv band), built ONCE ----
    float w9[KS];
#pragma unroll
    for (int k = 0; k < KS; ++k) w9[k] = w9g[k];

    v2f A[6];
#pragma unroll
    for (int t = 0; t < 6; ++t) {
        float a0, a1;
        {
            int j = t * 4 + 0 + 2 * hi;   // band column; this lane holds K = 0 + 2*hi
            int d = j - n;                // A[m][j] = w[j-m] if 0 <= j-m <= 8
            a0 = (d >= 0 && d < KS) ? w9[d] : 0.0f;
        }
        {
            int j = t * 4 + 1 + 2 * hi;   // K = 1 + 2*hi
            int d = j - n;
            a1 = (d >= 0 && d < KS) ? w9[d] : 0.0f;
        }
        // Opaque register barrier: prevents the register allocator from
        // rematerializing the cndmask select chains inside the 1279-step loop.
        asm volatile("" : "+v"(a0), "+v"(a1));
        A[t][0] = a0;
        A[t][1] = a1;
    }

    // loop-invariant LDS dword offsets for the 12 B-operand reads (pairs are contiguous)
    int boff[12];
#pragma unroll
    for (int t = 0; t < 6; ++t) {
        int e0 = wave_base + n * 16 + t * 4 + 2 * hi - 4;
        boff[2 * t + 0] = eslot(e0);
        boff[2 * t + 1] = eslot(e0 + 1);
    }
    const int s0 = eslot(h0);             // carry staging base (used by valid lanes only)

    const int w0  = (dir > 0) ? 0 : (WW - 1);
    const int h0c = (h0 <= HH - 8) ? h0 : (HH - 8);      // clamped: loads always in-bounds
    const float* sp = src + (size_t)h0c * WW;            // unpredicated load base
    float*       dp = dst + (size_t)h0  * WW;            // used only when valid

    // ---- first column: identity carry ----
    v8f carry;
#pragma unroll
    for (int r = 0; r < 8; ++r) carry[r] = sp[(size_t)r * WW + w0];

    // one scan step; cb is a compile-time-distinct colbuf half (kills per-step addr math)
    auto step = [&](float* __restrict__ cb, int i) {
        const int w = w0 + dir * i;

        // merged exec region: stage carry to LDS + write previous output column
        if (valid) {
            float4* p = (float4*)(cb + s0);
            p[0] = make_float4(carry[0], carry[1], carry[2], carry[3]);
            p[1] = make_float4(carry[4], carry[5], carry[6], carry[7]);
            const int wp = w - dir;
#pragma unroll
            for (int r = 0; r < 8; ++r) dp[(size_t)r * WW + wp] = carry[r];
        }
        __syncthreads();

        // prefetch this lane's 8 cache lines 32 columns ahead, once per 8 steps (uniform branch)
        if ((i & 7) == 0) {
            int wq = w + dir * 32;
            wq = wq < 0 ? 0 : (wq >= WW ? WW - 1 : wq);
#pragma unroll
            for (int r = 0; r < 8; ++r)
                __builtin_prefetch(&sp[(size_t)r * WW + wq], 0, 0);
        }

        // input column for this step (unpredicated; clamped base keeps it in-bounds)
        float xr[8];
#pragma unroll
        for (int r = 0; r < 8; ++r) xr[r] = sp[(size_t)r * WW + w];

        // conv = banded-matrix x carry-column via 6 chained fp32 WMMAs
        v8f conv = {0.f, 0.f, 0.f, 0.f, 0.f, 0.f, 0.f, 0.f};
#pragma unroll
        for (int t = 0; t < 6; ++t) {
            v2f Bv;
            Bv.x = cb[boff[2 * t + 0]];
            Bv.y = cb[boff[2 * t + 1]];
            conv = __builtin_amdgcn_wmma_f32_16x16x4_f32(
                /*neg_a=*/false, A[t], /*neg_b=*/false, Bv,
                /*c_mod=*/(short)0, conv, /*reuse_a=*/false, /*reuse_b=*/false);
        }

        // carry = x + relu(conv)  (single v_max_num per element)
#pragma unroll
        for (int r = 0; r < 8; ++r)
            carry[r] = xr[r] + __builtin_fmaxf(conv[r], 0.0f);
    };

    // manual unroll-by-2: buffer parity is compile-time within each body
    step(cb1, 1);
    for (int i = 2; i < WW; i += 2) {
        step(cb0, i);
        step(cb1, i + 1);
    }

    // flush final column
    if (valid) {
        const int wl = w0 + dir * (WW - 1);
#pragma unroll
        for (int r = 0; r < 8; ++r) dp[(size_t)r * WW + wl] = carry[r];
    }
}

__global__ __launch_bounds__(64)
void scnn_bidir_kernel(const float* __restrict__ x,
                       const float* __restrict__ w_lr,
                       const float* __restrict__ w_rl,
                       float* __restrict__ out)
{
    __shared__ float colbuf[2][COLBUF_DW];

    const int tid  = threadIdx.x;
    const int lane = tid & 31;
    const int wave = tid >> 5;           // 0: h 0..255, 1: h 256..383
    const int n    = lane & 15;          // N (h-block) index within tile
    const int hi   = lane >> 4;          // half-wave selector
    const int wave_base = wave * 256;
    const int h0   = wave_base + n * 16 + 8 * hi;   // first of this lane's 8 consecutive h
    const bool valid = (h0 < HH);

    // zero LDS: halo slots must read 0 forever; live slots are always overwritten
    for (int i = tid; i < 2 * COLBUF_DW; i += 64) ((float*)colbuf)[i] = 0.0f;
    __syncthreads();

    const int b = blockIdx.x;
    const float* xb = x   + (size_t)b * HH * WW;
    float*       ob = out + (size_t)b * HH * WW;

    // pass 1: x -> out (left to right, w_lr)
    run_pass(xb, ob, w_lr, colbuf[0], colbuf[1], wave_base, n, hi, h0, valid, +1);
    __syncthreads();
    // pass 2: out -> out in place (right to left, w_rl); each lane re-reads only its own
    // writes (same-wave same-address ordering + data deps), and column w is read before
    // the following step overwrites it.
    run_pass(ob, ob, w_rl, colbuf[0], colbuf[1], wave_base, n, hi, h0, valid, -1);
}

extern "C" void kernel_launch(void* const* d_in, const int* in_sizes, int n_in,
                              void* d_out, int out_size, void* d_ws, size_t ws_size,
                              hipStream_t stream)
{
    const float* x    = (const float*)d_in[0];
    const float* w_lr = (const float*)d_in[1];
    const float* w_rl = (const float*)d_in[2];
    float* out        = (float*)d_out;
    scnn_bidir_kernel<<<BATCH, 64, 0, stream>>>(x, w_lr, w_rl, out);
}